// DiversificationBlock_50861002719974
// MI455X (gfx1250) — compile-verified
//
#include <hip/hip_runtime.h>
#include <stdint.h>

#define ALPHA_F   0.1f
#define CH_ELEMS  784              // 28*28
#define CHUNKS    196              // 784/4 float4 chunks
#define NPATCH    16               // 4x4 grid of 7x7 patches
#define BLOCK     224              // 7 waves of 32; >= CHUNKS

typedef float v4f __attribute__((ext_vector_type(4)));

// Order-preserving float<->uint key (works for LDS u32 atomics)
__device__ __forceinline__ unsigned fkey(float f) {
  unsigned b = __float_as_uint(f);
  return (b & 0x80000000u) ? ~b : (b | 0x80000000u);
}
__device__ __forceinline__ float funkey(unsigned k) {
  unsigned b = (k & 0x80000000u) ? (k ^ 0x80000000u) : ~k;
  return __uint_as_float(b);
}

__global__ __launch_bounds__(BLOCK)
void divblock_kernel(const float* __restrict__ cam,
                     const float* __restrict__ u_peak,
                     const float* __restrict__ u_patch,
                     float* __restrict__ out) {
  __shared__ unsigned smin[NPATCH];
  __shared__ unsigned smax[NPATCH];
  __shared__ unsigned sflag;
  __shared__ float    scam[CH_ELEMS];

  const int    tid = threadIdx.x;
  const size_t ch  = blockIdx.x;
  const float* src = cam + ch * CH_ELEMS;
  float*       dst = out + ch * CH_ELEMS;

  if (tid < NPATCH) { smin[tid] = 0xFFFFFFFFu; smax[tid] = 0u; }
  if (tid == 0)     { sflag = 0u; }

  const bool active = (tid < CHUNKS);
  v4f v = (v4f){0.f, 0.f, 0.f, 0.f};

  // --- CDNA5 async copy: global -> LDS, 16B per lane, tracked by ASYNCcnt ---
  if (active) {
    unsigned     lds_off = (unsigned)(uintptr_t)(&scam[tid * 4]);
    const float* gaddr   = src + tid * 4;
    asm volatile("global_load_async_to_lds_b128 %0, %1, off"
                 :: "v"(lds_off), "v"(gaddr) : "memory");
  }
  asm volatile("s_wait_asynccnt 0" ::: "memory");
  if (active) {
    v = *(const v4f*)(&scam[tid * 4]);  // ds_load_b128, own chunk only
  }

  __syncthreads();  // smin/smax init visible before atomics

  if (active) {
    // chunk t covers row = t/7, cols [4*(t%7) .. +3]; 7 chunks per row
    const int q   = tid % 7;
    const int row = tid / 7;
    const int pr4 = (row / 7) * 4;
    const int c0  = q * 4;

    float xs[4] = {v.x, v.y, v.z, v.w};
    unsigned curp = (unsigned)(pr4 + c0 / 7);
    float fm = xs[0], fM = xs[0];
#pragma unroll
    for (int i = 1; i < 4; ++i) {
      unsigned p = (unsigned)(pr4 + (c0 + i) / 7);
      if (p == curp) {
        fm = fminf(fm, xs[i]);
        fM = fmaxf(fM, xs[i]);
      } else {  // chunk straddles a patch boundary: flush and restart
        atomicMin(&smin[curp], fkey(fm));
        atomicMax(&smax[curp], fkey(fM));
        curp = p; fm = xs[i]; fM = xs[i];
      }
    }
    atomicMin(&smin[curp], fkey(fm));
    atomicMax(&smax[curp], fkey(fM));
  }
  __syncthreads();  // patch extrema complete

  if (tid < NPATCH) {
    unsigned mnk = smin[0], mxk = smax[0];
#pragma unroll
    for (int i = 1; i < NPATCH; ++i) {
      mnk = min(mnk, smin[i]);
      mxk = max(mxk, smax[i]);
    }
    const float mn = funkey(mnk);
    const float mx = funkey(mxk);
    if (mx > mn) {  // else: reference normalizes to NaN -> never flags
      const float denom = mx - mn;
      const float apmax = (funkey(smax[tid]) - mn) / denom;  // max of a in patch
      const float apmin = (funkey(smin[tid]) - mn) / denom;  // min of a in patch
      const bool  sel     = u_patch[ch * NPATCH + tid] < apmax;
      const bool  nonpeak = !(apmin == 1.0f);  // patch has a non-peak element
      unsigned f = (sel && nonpeak) ? 1u : 0u;
      if (tid == 0 && u_peak[ch] < 1.0f) f |= 1u;  // rc * peak_mask term
      if (f) atomicOr(&sflag, 1u);
    }
  }
  __syncthreads();  // flag final

  if (active) {
    const float s = sflag ? ALPHA_F : 1.0f;
    v.x *= s; v.y *= s; v.z *= s; v.w *= s;
    __builtin_nontemporal_store(v, (v4f*)(dst + tid * 4));  // stream out, NT
  }
}

extern "C" void kernel_launch(void* const* d_in, const int* in_sizes, int n_in,
                              void* d_out, int out_size, void* d_ws, size_t ws_size,
                              hipStream_t stream) {
  const float* cam     = (const float*)d_in[0];
  const float* u_peak  = (const float*)d_in[1];
  const float* u_patch = (const float*)d_in[2];
  float*       out     = (float*)d_out;

  const int nch = in_sizes[1];  // B*C = 131072 channels
  divblock_kernel<<<dim3((unsigned)nch), dim3(BLOCK), 0, stream>>>(
      cam, u_peak, u_patch, out);
}